// SpikingConv3DLayer_4853313044929
// MI455X (gfx1250) — compile-verified
//
#include <hip/hip_runtime.h>
#include <cstddef>

typedef __attribute__((ext_vector_type(2))) float v2f;
typedef __attribute__((ext_vector_type(8))) float v8f;

#define B_N   8
#define C_IN  2
#define T_N   64
#define H_N   64
#define W_N   64
#define C_OUT 32
#define KVOL  54          // C_IN * 27 conv taps
#define NKC   15          // K permuted/padded to 60: 3 dt-groups of 20 (18 real + 2 pad)
#define NKD   8           // 32/4 K-chunks for the 32x32 d-matrix GEMM
#define HW    (H_N*W_N)

#define ALPHA_F 0.8187307530779818f   // exp(-0.001/0.005)
#define BETA_F  0.9048374180359595f   // exp(-0.001/0.010)
#define EPS_F   1e-8f

#define PLANE      396                // C_IN(2) * rows(3) * halo cols(66) floats
#define SLOT_BYTES (PLANE*4)
#define WG_THREADS 128

// D = A(16x4) * B(4x16) + C, full fp32 (CDNA5 V_WMMA_F32_16X16X4_F32)
__device__ __forceinline__ v8f wmma_f32(v2f a, v2f b, v8f c) {
  return __builtin_amdgcn_wmma_f32_16x16x4_f32(false, a, false, b, (short)0, c,
                                               false, false);
}

// ---- setup: gram matrix d[o][p] = <w_o, w_p>, inv_norm[o] = 1/(d[o][o]+eps) ----
__global__ void gram_kernel(const float* __restrict__ w,
                            float* __restrict__ dmat,
                            float* __restrict__ invn) {
  int tid = threadIdx.x;          // 1024 threads
  int o = tid >> 5;
  int p = tid & 31;
  float s = 0.f;
  #pragma unroll 6
  for (int k = 0; k < KVOL; ++k) s += w[o*KVOL + k] * w[p*KVOL + k];
  dmat[o*32 + p] = s;
  if (o == p) invn[o] = 1.f / (s + EPS_F);
}

// load one x time-plane (3 h-rows with w halo, both input channels) into an LDS ring slot
__device__ __forceinline__ void load_plane(float* xs, int slot,
                                           const float* __restrict__ x,
                                           int bIdx, int h, int tt, int tid) {
  float* dst = xs + slot * PLANE;
  #pragma unroll
  for (int i = 0; i < 4; ++i) {                 // 4*128 >= 396
    int idx = tid + i * WG_THREADS;
    if (idx < PLANE) {
      int ci  = idx / 198;
      int rem = idx - ci * 198;
      int row = rem / 66;                       // 0..2 -> h-1..h+1
      int col = rem - row * 66;                 // 0..65 -> w-1..64
      int hh = h - 1 + row;
      int ww = col - 1;
      float v = 0.f;
      if (tt < T_N && (unsigned)hh < (unsigned)H_N && (unsigned)ww < (unsigned)W_N)
        v = x[(((size_t)bIdx * C_IN + ci) * T_N + tt) * HW + (size_t)hh * W_N + ww];
      dst[idx] = v;
    }
  }
}

__global__ __launch_bounds__(WG_THREADS)
void snn_step_kernel(const float* __restrict__ x,
                     const float* __restrict__ w,
                     const float* __restrict__ bvec,
                     const float* __restrict__ dmat,
                     const float* __restrict__ invn,
                     float* __restrict__ out) {
  __shared__ __align__(16) float xs[4 * PLANE];        // 4-slot ring of x time-planes
  __shared__ __align__(16) float spk_lds[C_OUT * W_N]; // spikes [c][w]

  const int tid   = threadIdx.x;
  const int lane  = tid & 31;
  const int wave  = tid >> 5;
  const int n     = lane & 15;               // WMMA N / M row index
  const int khalf = lane >> 4;               // which K pair this half-wave owns
  const int wcol  = wave * 16 + n;           // output w position (0..63)

  const int bIdx = blockIdx.x >> 6;
  const int h    = blockIdx.x & 63;

  // zero spike carry (t = -1)
  for (int i = tid; i < C_OUT * W_N; i += WG_THREADS) spk_lds[i] = 0.f;

  // K permutation for the conv GEMM: chunk kk has dt = kk/5 (compile-time);
  // within a dt-group, tap q = (kk%5)*4 + khalf*2 + j, q<18 real: (ci,dh,dw).
  // Original w flat index for a tap = ci*27 + dt*9 + dh*3 + dw.

  // ---- resident A fragments: permuted conv weights ----
  v2f aconv[2][NKC];
  #pragma unroll
  for (int mt = 0; mt < 2; ++mt) {
    #pragma unroll
    for (int kk = 0; kk < NKC; ++kk) {
      const int dt = kk / 5;
      const int m  = mt * 16 + n;
      float av[2];
      #pragma unroll
      for (int j = 0; j < 2; ++j) {
        int q = (kk % 5) * 4 + khalf * 2 + j;
        float v = 0.f;
        if (q < 18) {
          int ci = q / 9;  int rem = q - ci * 9;
          int dh = rem / 3; int dw = rem - dh * 3;
          v = w[m * KVOL + ci * 27 + dt * 9 + dh * 3 + dw];
        }
        av[j] = v;
      }
      aconv[mt][kk].x = av[0];
      aconv[mt][kk].y = av[1];
    }
  }

  // ---- resident A fragments: d^T (d is symmetric), A[m][k] = d[k][m] ----
  v2f adf[2][NKD];
  #pragma unroll
  for (int mt = 0; mt < 2; ++mt) {
    #pragma unroll
    for (int kk = 0; kk < NKD; ++kk) {
      int m  = mt * 16 + n;
      int k0 = kk * 4 + khalf * 2;
      adf[mt][kk].x = dmat[(k0    ) * 32 + m];
      adf[mt][kk].y = dmat[(k0 + 1) * 32 + m];
    }
  }

  // per-lane inv_norm / bias in C-layout: element (mt, r) has m = mt*16 + khalf*8 + r
  float invnr[2][8], biasr[2][8];
  #pragma unroll
  for (int mt = 0; mt < 2; ++mt)
    #pragma unroll
    for (int r = 0; r < 8; ++r) {
      int m = mt * 16 + khalf * 8 + r;
      invnr[mt][r] = invn[m];
      biasr[mt][r] = bvec[m];
    }

  // ---- static per-lane LDS byte offsets for the im2col gather (dt excluded) ----
  int soff[NKC][2];
  #pragma unroll
  for (int kk = 0; kk < NKC; ++kk) {
    #pragma unroll
    for (int j = 0; j < 2; ++j) {
      int q  = (kk % 5) * 4 + khalf * 2 + j;
      int qc = (q < 18) ? q : 0;
      int ci = qc / 9;   int rem = qc - ci * 9;
      int dh = rem / 3;  int dw = rem - dh * 3;
      soff[kk][j] = (ci * 198 + dh * 66 + wcol + dw) * 4;
    }
  }
  const float maskTail = (khalf == 0) ? 1.f : 0.f;  // q=18,19 padding lanes

  // states in WMMA C-layout
  v8f memv[2] = {{0,0,0,0,0,0,0,0}, {0,0,0,0,0,0,0,0}};
  v8f synv[2] = {{0,0,0,0,0,0,0,0}, {0,0,0,0,0,0,0,0}};

  // prime the plane ring: slot3 = zeros (t = -1), slot0 = plane 0
  { float* dst = xs + 3 * PLANE;
    for (int i = tid; i < PLANE; i += WG_THREADS) dst[i] = 0.f; }
  load_plane(xs, 0, x, bIdx, h, 0, tid);

  for (int t = 0; t < T_N; ++t) {
    load_plane(xs, (t + 1) & 3, x, bIdx, h, t + 1, tid);
    __syncthreads();   // plane t+1 visible; slot overwritten above last read at t-2

    // wave-uniform ring-slot bases for dt = 0,1,2 (scalar)
    int sb[3];
    sb[0] = ((t + 3) & 3) * SLOT_BYTES;   // tt = t-1
    sb[1] = ((t    ) & 3) * SLOT_BYTES;   // tt = t
    sb[2] = ((t + 1) & 3) * SLOT_BYTES;   // tt = t+1

    // ---- conv GEMM: acc[mt] = W(16x60) * im2col(60x16) ----
    v8f acc[2] = {{0,0,0,0,0,0,0,0}, {0,0,0,0,0,0,0,0}};
    #pragma unroll
    for (int kk = 0; kk < NKC; ++kk) {
      const int base = sb[kk / 5];        // compile-time selection after unroll
      int a0 = soff[kk][0] + base;
      int a1 = soff[kk][1] + base;
      v2f bb;
      bb.x = *(const float*)((const char*)xs + a0);
      bb.y = *(const float*)((const char*)xs + a1);
      if ((kk % 5) == 4) { bb.x *= maskTail; bb.y *= maskTail; }
      acc[0] = wmma_f32(aconv[0][kk], bb, acc[0]);
      acc[1] = wmma_f32(aconv[1][kk], bb, acc[1]);
    }

    // ---- rst GEMM: rst[mt] = d^T(16x32) * spk(32x16), spikes of t-1 from LDS ----
    v8f rst[2] = {{0,0,0,0,0,0,0,0}, {0,0,0,0,0,0,0,0}};
    #pragma unroll
    for (int kk = 0; kk < NKD; ++kk) {
      int c0 = kk * 4 + khalf * 2;
      v2f sbv;
      sbv.x = spk_lds[(c0    ) * W_N + wcol];
      sbv.y = spk_lds[(c0 + 1) * W_N + wcol];
      rst[0] = wmma_f32(adf[0][kk], sbv, rst[0]);
      rst[1] = wmma_f32(adf[1][kk], sbv, rst[1]);
    }

    // ---- LIF state update + spike into LDS ----
    #pragma unroll
    for (int mt = 0; mt < 2; ++mt) {
      #pragma unroll
      for (int r = 0; r < 8; ++r) {
        float oldsyn = synv[mt][r];
        float nsyn = ALPHA_F * oldsyn + acc[mt][r];
        float nmem = BETA_F * memv[mt][r] + oldsyn - rst[mt][r];
        float mthr = nmem * invnr[mt][r] - biasr[mt][r];
        float s = (mthr > 0.f) ? 1.f : 0.f;
        synv[mt][r] = nsyn;
        memv[mt][r] = nmem;
        int m = mt * 16 + khalf * 8 + r;
        spk_lds[m * W_N + wcol] = s;     // carry for next step + staging for store
      }
    }
    __syncthreads();   // full 32x64 spike tile visible to all waves

    // ---- cooperative, fully-coalesced output write: 128 threads x 16 floats ----
    {
      const int c   = tid >> 2;          // cout 0..31
      const int wsg = (tid & 3) * 16;    // 0,16,32,48
      const float4* src = (const float4*)(spk_lds + c * W_N + wsg);
      float4* dst = (float4*)(out + (((size_t)(bIdx * C_OUT + c) * T_N + t) * HW)
                                  + (size_t)h * W_N + wsg);
      dst[0] = src[0];
      dst[1] = src[1];
      dst[2] = src[2];
      dst[3] = src[3];
    }
    // next iteration's spk_lds writes are separated from these reads by the
    // top-of-loop __syncthreads()
  }
}

extern "C" void kernel_launch(void* const* d_in, const int* in_sizes, int n_in,
                              void* d_out, int out_size, void* d_ws, size_t ws_size,
                              hipStream_t stream) {
  (void)in_sizes; (void)n_in; (void)out_size; (void)ws_size;
  const float* x = (const float*)d_in[0];
  const float* w = (const float*)d_in[1];
  const float* b = (const float*)d_in[2];
  float* out  = (float*)d_out;
  float* dmat = (float*)d_ws;                 // 32*32 floats
  float* invn = dmat + C_OUT * C_OUT;         // 32 floats

  gram_kernel<<<1, 1024, 0, stream>>>(w, dmat, invn);
  snn_step_kernel<<<B_N * H_N, WG_THREADS, 0, stream>>>(x, w, b, dmat, invn, out);
}